// UniversalGRU_15169824489765
// MI455X (gfx1250) — compile-verified
//
#include <hip/hip_runtime.h>

#define H   64
#define G   192
#define TT  512
#define BATCH 2048
#define BT  16      // batch tile (WMMA M)
#define WST 68      // padded LDS row stride (floats) -> conflict-free banks

typedef __attribute__((ext_vector_type(2))) float v2f;
typedef __attribute__((ext_vector_type(8))) float v8f;

__device__ __forceinline__ float sig_(float x)  { return 1.0f / (1.0f + __expf(-x)); }
__device__ __forceinline__ float tanh_(float x) { return 1.0f - 2.0f / (1.0f + __expf(2.0f * x)); }

// Accumulate acc += A(16x64) * B(64x16) using 16 chained V_WMMA_F32_16X16X4_F32.
// B fragment: lane n = lane&15, vgpr0 = B[2*hi][n], vgpr1 = B[2*hi+1][n];
// weight array is [192][WST] row-major, row g = gbase + n, cols = k.
__device__ __forceinline__ v8f wmma_k64(v8f acc, const v2f* Af,
                                        const float* __restrict__ w,
                                        int grow, int hi2) {
#pragma unroll
  for (int k = 0; k < 16; ++k) {
    v2f b = *(const v2f*)(w + grow + 4 * k + hi2);
    acc = __builtin_amdgcn_wmma_f32_16x16x4_f32(false, Af[k], false, b,
                                                (short)0, acc, false, false);
  }
  return acc;
}

// A fragment k: lane m = lane&15, float2 at h[m][4k + 2*hi]
__device__ __forceinline__ void load_afrags(v2f* Af, const float* h, int m, int hi2) {
#pragma unroll
  for (int k = 0; k < 16; ++k)
    Af[k] = *(const v2f*)(h + m * WST + 4 * k + hi2);
}

__global__ __launch_bounds__(128) void gru2_fused_kernel(
    const float* __restrict__ x,
    const float* __restrict__ Wih0, const float* __restrict__ Whh0,
    const float* __restrict__ bih0, const float* __restrict__ bhh0,
    const float* __restrict__ Wih1, const float* __restrict__ Whh1,
    const float* __restrict__ bih1, const float* __restrict__ bhh1,
    const float* __restrict__ Wfc,  const float* __restrict__ bfc,
    float* __restrict__ out)
{
  extern __shared__ float smem[];
  float* s_whh0 = smem;                    // G*WST
  float* s_wih1 = s_whh0 + G * WST;        // G*WST
  float* s_whh1 = s_wih1 + G * WST;        // G*WST
  float* s_h0   = s_whh1 + G * WST;        // 2 * BT*WST (ping-pong)
  float* s_h1   = s_h0 + 2 * BT * WST;     // 2 * BT*WST (ping-pong)
  float* s_xs   = s_h1 + 2 * BT * WST;     // 2 * BT     (ping-pong)

  const int tid    = threadIdx.x;
  const int wave   = tid >> 5;             // 0..3 : N-split across waves
  const int lane   = tid & 31;
  const int lane16 = lane & 15;
  const int hi     = lane >> 4;
  const int hi2    = 2 * hi;
  const int base   = blockIdx.x * BT;

  // Stage weights ([192][64] row-major -> padded stride) and zero state.
  for (int idx = tid; idx < G * H; idx += 128) {
    int g = idx >> 6, k = idx & 63;
    s_whh0[g * WST + k] = Whh0[idx];
    s_wih1[g * WST + k] = Wih1[idx];
    s_whh1[g * WST + k] = Whh1[idx];
  }
  for (int idx = tid; idx < 2 * BT * WST; idx += 128) { s_h0[idx] = 0.f; s_h1[idx] = 0.f; }
  if (tid < BT) s_xs[tid] = x[(base + tid) * TT + 0];  // xs for t=0 -> buffer 0

  // Per-lane gate rows: tile slot j -> gate g_j (r: j=0, z: j=1, n: j=2)
  const int g0 = 0 * H + 16 * wave + lane16;
  const int g1 = 1 * H + 16 * wave + lane16;
  const int g2 = 2 * H + 16 * wave + lane16;
  const float w0r = Wih0[g0], w0z = Wih0[g1], w0n = Wih0[g2];
  const float c0r = bih0[g0] + bhh0[g0], c0z = bih0[g1] + bhh0[g1];
  const float b0nh = bhh0[g2], b0nx = bih0[g2];
  const float c1r = bih1[g0] + bhh1[g0], c1z = bih1[g1] + bhh1[g1];
  const float b1nh = bhh1[g2], b1nx = bih1[g2];
  const int col = 16 * wave + lane16;      // hidden column owned by this lane
  const int mA  = lane16;

  __syncthreads();

  v2f Af[16];

  for (int t = 0; t < TT; ++t) {
    const int p   = t & 1;                 // current state/xs buffer
    const int q   = p ^ 1;                 // next state/xs buffer
    float* h0c = s_h0 + p * BT * WST;
    float* h0n = s_h0 + q * BT * WST;
    float* h1c = s_h1 + p * BT * WST;
    float* h1n = s_h1 + q * BT * WST;

    // Prefetch x_{t+1} early; latency hides behind this step's WMMA work.
    const int tn = (t + 1 < TT) ? (t + 1) : (TT - 1);
    float xnext = (tid < BT) ? x[(base + tid) * TT + tn] : 0.0f;

    __syncthreads();   // S0: prev-step h writes (into p-buffers) + xs[p] visible

    float xsr[8];
#pragma unroll
    for (int i = 0; i < 8; ++i) xsr[i] = s_xs[p * BT + i + 8 * hi];

    // ---- layer 0: hp0 = h0 @ Whh0^T  (x-proj + biases folded into init) ----
    load_afrags(Af, h0c, mA, hi2);
    v8f a0r, a0z, a0n;
#pragma unroll
    for (int i = 0; i < 8; ++i) {
      a0r[i] = xsr[i] * w0r + c0r;
      a0z[i] = xsr[i] * w0z + c0z;
      a0n[i] = b0nh;
    }
    a0r = wmma_k64(a0r, Af, s_whh0, g0 * WST, hi2);
    a0z = wmma_k64(a0z, Af, s_whh0, g1 * WST, hi2);
    a0n = wmma_k64(a0n, Af, s_whh0, g2 * WST, hi2);

    // ---- layer 1 recurrent part: h1 @ Whh1^T ----
    load_afrags(Af, h1c, mA, hi2);
    v8f a1r, a1z, a1nh;
#pragma unroll
    for (int i = 0; i < 8; ++i) { a1r[i] = c1r; a1z[i] = c1z; a1nh[i] = b1nh; }
    a1r  = wmma_k64(a1r,  Af, s_whh1, g0 * WST, hi2);
    a1z  = wmma_k64(a1z,  Af, s_whh1, g1 * WST, hi2);
    a1nh = wmma_k64(a1nh, Af, s_whh1, g2 * WST, hi2);

    // ---- layer 0 gates -> h0_new, written to the OTHER buffer (no race) ----
#pragma unroll
    for (int i = 0; i < 8; ++i) {
      float r  = sig_(a0r[i]);
      float z  = sig_(a0z[i]);
      float xn = xsr[i] * w0n + b0nx;
      float n  = tanh_(xn + r * a0n[i]);
      float hp = h0c[(i + 8 * hi) * WST + col];
      h0n[(i + 8 * hi) * WST + col] = (1.0f - z) * n + z * hp;
    }
    __syncthreads();   // S1: h0_new (q-buffer) visible to all waves

    // ---- layer 1 input part: h0_new @ Wih1^T, accumulated into same regs ----
    load_afrags(Af, h0n, mA, hi2);
    v8f a1nx;
#pragma unroll
    for (int i = 0; i < 8; ++i) a1nx[i] = b1nx;
    a1r  = wmma_k64(a1r,  Af, s_wih1, g0 * WST, hi2);
    a1z  = wmma_k64(a1z,  Af, s_wih1, g1 * WST, hi2);
    a1nx = wmma_k64(a1nx, Af, s_wih1, g2 * WST, hi2);

    // ---- layer 1 gates -> h1_new (other buffer) ----
#pragma unroll
    for (int i = 0; i < 8; ++i) {
      float r  = sig_(a1r[i]);
      float z  = sig_(a1z[i]);
      float n  = tanh_(a1nx[i] + r * a1nh[i]);
      float hp = h1c[(i + 8 * hi) * WST + col];
      h1n[(i + 8 * hi) * WST + col] = (1.0f - z) * n + z * hp;
    }

    // Commit prefetched x_{t+1} into the next xs buffer (guarded by next S0).
    if (tid < BT) s_xs[q * BT + tid] = xnext;
    // next loop-top S0 fences all q-buffer writes before they are read
  }

  __syncthreads();
  {
    // Final h1 state lives in buffer written at t = TT-1: index ((TT-1)&1)^1.
    const float* hf = s_h1 + ((((TT - 1) & 1) ^ 1)) * BT * WST;
    if (tid < BT) {
      float acc = bfc[0];
#pragma unroll 4
      for (int k = 0; k < H; ++k) acc += hf[tid * WST + k] * Wfc[k];
      out[base + tid] = acc;
    }
  }
}

extern "C" void kernel_launch(void* const* d_in, const int* in_sizes, int n_in,
                              void* d_out, int out_size, void* d_ws, size_t ws_size,
                              hipStream_t stream) {
  (void)in_sizes; (void)n_in; (void)out_size; (void)d_ws; (void)ws_size;
  const float* x    = (const float*)d_in[0];
  const float* Wih0 = (const float*)d_in[1];
  const float* Whh0 = (const float*)d_in[2];
  const float* bih0 = (const float*)d_in[3];
  const float* bhh0 = (const float*)d_in[4];
  const float* Wih1 = (const float*)d_in[5];
  const float* Whh1 = (const float*)d_in[6];
  const float* bih1 = (const float*)d_in[7];
  const float* bhh1 = (const float*)d_in[8];
  const float* Wfc  = (const float*)d_in[9];
  const float* bfc  = (const float*)d_in[10];
  float* out = (float*)d_out;

  const size_t lds_bytes =
      (size_t)(3 * G * WST + 4 * BT * WST + 2 * BT) * sizeof(float);
  dim3 grid(BATCH / BT);   // 128 workgroups, one 16-row batch tile each
  dim3 block(128);         // 4 waves, N-split of the 192 gate columns
  gru2_fused_kernel<<<grid, block, lds_bytes, stream>>>(
      x, Wih0, Whh0, bih0, bhh0, Wih1, Whh1, bih1, bhh1, Wfc, bfc, out);
}